// SAG_214748365118
// MI455X (gfx1250) — compile-verified
//
#include <hip/hip_runtime.h>
#include <hip/hip_bf16.h>
#include <math.h>

// ---------------- problem constants (match reference) ----------------
#define BGR   128            // graphs
#define NPG   512            // nodes per graph (layer 0)
#define EPG   8192
#define ETOT  (BGR*EPG)      // 1,048,576 edges
#define HDIM  128            // DIN == H == 128
#define DOUT  10
#define K1    410
#define K2    328
#define N0    (BGR*NPG)      // 65536
#define N1    (BGR*K1)       // 52480
#define N2    (BGR*K2)       // 41984
#define BNEPS 1e-5f
#define LWS   132            // padded LDS stride (floats) -> conflict-free ds_load_b64

typedef __attribute__((ext_vector_type(2))) float v2f;
typedef __attribute__((ext_vector_type(8))) float v8f;

// ---------------- WMMA GEMM: C[nrows x 128] = A[nrows x 128] @ W[128 x 128] ----
// W is row-major [k][n]. Staged TRANSPOSED in LDS as lw[n*LWS + k] so each
// B-fragment (two consecutive K values of one column) is a single aligned
// ds_load_b64; stride 132 interleaves the two K-halves across all 64 banks.
// One 256-thread block = 8 waves; wave w computes rows [blk*128 + w*16, +16).
__global__ __launch_bounds__(256) void gemm_wmma_f32(const float* __restrict__ A,
                                                     const float* __restrict__ W,
                                                     float* __restrict__ C) {
  __shared__ float lw[128 * LWS];                      // ~67.5 KB of 320 KB WGP LDS
  for (int i = threadIdx.x; i < 128 * 128; i += blockDim.x) {
    int k = i >> 7, n = i & 127;
    lw[n * LWS + k] = W[i];                            // coalesced global read
  }
  __syncthreads();

  const int wave = threadIdx.x >> 5;
  const int lane = threadIdx.x & 31;
  const int half = lane >> 4;          // 0: K%4 in {0,1}, 1: K%4 in {2,3}
  const int lidx = lane & 15;          // M (for A) / N-within-tile (for B, C/D)
  const int rowbase = blockIdx.x * 128 + wave * 16;

  v8f acc[8];
  for (int j = 0; j < 8; ++j)
    for (int i = 0; i < 8; ++i) acc[j][i] = 0.0f;

  const float* arow = A + (size_t)(rowbase + lidx) * HDIM + half * 2;
  for (int k = 0; k < 128; k += 4) {
    // A 16x4 frag: VGPR0 = K{k | k+2}, VGPR1 = K{k+1 | k+3}  (one global b64)
    v2f a = *(const v2f*)(arow + k);
    const int kk = k + 2 * half;
    for (int j = 0; j < 8; ++j) {
      // B 4x16 frag, mirrored layout: one aligned ds_load_b64
      v2f b = *(const v2f*)(lw + (j * 16 + lidx) * LWS + kk);
      acc[j] = __builtin_amdgcn_wmma_f32_16x16x4_f32(
          false, a, false, b, (short)0, acc[j], false, false);
    }
  }
  // C/D layout: VGPR i -> row i (lanes 0-15) / row i+8 (lanes 16-31), N = lidx
  for (int j = 0; j < 8; ++j)
    for (int i = 0; i < 8; ++i) {
      const int row = rowbase + i + half * 8;
      C[(size_t)row * HDIM + j * 16 + lidx] = acc[j][i];
    }
}

// ---------------- small helper kernels ----------------
__global__ void fill_f32(float* p, float v, long long n) {
  long long t = (long long)blockIdx.x * blockDim.x + threadIdx.x;
  if (t < n) p[t] = v;
}

__global__ void deg_edges(const int* __restrict__ dst, const float* __restrict__ em,
                          float* __restrict__ deg, int ne) {
  int e = blockIdx.x * blockDim.x + threadIdx.x;
  if (e >= ne) return;
  float m = em ? em[e] : 1.0f;
  if (m != 0.0f) atomicAdd(&deg[dst[e]], m);
}

__global__ void make_dinv(const float* __restrict__ deg, float* __restrict__ dinv, int n) {
  int i = blockIdx.x * blockDim.x + threadIdx.x;
  if (i < n) dinv[i] = rsqrtf(deg[i]);
}

// one thread per (edge, feature): agg[dst] += hw[src] * dinv[src]*dinv[dst]*em
__global__ void edge_agg(const int* __restrict__ src, const int* __restrict__ dst,
                         const float* __restrict__ em, const float* __restrict__ dinv,
                         const float* __restrict__ hw, float* __restrict__ agg, int ne) {
  long long t = (long long)blockIdx.x * blockDim.x + threadIdx.x;
  if (t >= (long long)ne * HDIM) return;
  int e = (int)(t >> 7), f = (int)(t & 127);
  int s = src[e], d = dst[e];
  float m = em ? em[e] : 1.0f;
  float coef = dinv[s] * dinv[d] * m;
  if (coef != 0.0f)
    atomicAdd(&agg[(size_t)d * HDIM + f], hw[(size_t)s * HDIM + f] * coef);
}

// h = agg(in place) + hw * dinv^2 + bias
__global__ void combine(float* __restrict__ h, const float* __restrict__ hw,
                        const float* __restrict__ dinv, const float* __restrict__ bias,
                        long long n128) {
  long long t = (long long)blockIdx.x * blockDim.x + threadIdx.x;
  if (t >= n128) return;
  int node = (int)(t >> 7), f = (int)(t & 127);
  float d = dinv[node];
  h[t] = h[t] + hw[t] * d * d + bias[f];
}

// per-node dot(h_row, w) with one wave per node
__global__ void score_gemv(const float* __restrict__ h, const float* __restrict__ w,
                           float* __restrict__ out, int n) {
  int wv = (blockIdx.x * blockDim.x + threadIdx.x) >> 5;
  int lane = threadIdx.x & 31;
  if (wv >= n) return;
  const float* row = h + (size_t)wv * HDIM;
  float s = 0.0f;
  for (int f = lane; f < HDIM; f += 32) s += row[f] * w[f];
  for (int o = 16; o > 0; o >>= 1) s += __shfl_down(s, o, 32);
  if (lane == 0) out[wv] = s;
}

__global__ void score_edge_agg(const int* __restrict__ src, const int* __restrict__ dst,
                               const float* __restrict__ em, const float* __restrict__ dinv,
                               const float* __restrict__ shw, float* __restrict__ sagg, int ne) {
  int e = blockIdx.x * blockDim.x + threadIdx.x;
  if (e >= ne) return;
  int s = src[e], d = dst[e];
  float m = em ? em[e] : 1.0f;
  float coef = dinv[s] * dinv[d] * m;
  if (coef != 0.0f) atomicAdd(&sagg[d], shw[s] * coef);
}

__global__ void score_combine(float* __restrict__ sc, const float* __restrict__ sagg,
                              const float* __restrict__ shw, const float* __restrict__ dinv,
                              const float* __restrict__ sb, int n) {
  int i = blockIdx.x * blockDim.x + threadIdx.x;
  if (i >= n) return;
  float d = dinv[i];
  sc[i] = sagg[i] + shw[i] * d * d + sb[0];
}

// deterministic per-graph top-k selection; keeps the k best-ranked nodes in
// original-index order (output invariant: all downstream ops are permutation
// equivariant/invariant). remap[old_global] = new_global or -1.
__global__ __launch_bounds__(512) void pool_select(const float* __restrict__ sc,
                                                   int n_per, int k, int* __restrict__ remap) {
  __shared__ float ls[512];
  __shared__ int lrank[512];
  int g = blockIdx.x, i = threadIdx.x;
  if (i < n_per) ls[i] = sc[g * n_per + i];
  __syncthreads();
  int rank = 0;
  if (i < n_per) {
    float si = ls[i];
    for (int j = 0; j < n_per; ++j) {
      float sj = ls[j];
      rank += (sj > si) || (sj == si && j < i);
    }
    lrank[i] = rank;
  }
  __syncthreads();
  if (i < n_per) {
    int newidx = 0;
    for (int j = 0; j < i; ++j) newidx += (lrank[j] < k);
    remap[g * n_per + i] = (rank < k) ? (g * k + newidx) : -1;
  }
}

// h_out[remap[old]] = h_in[old] * tanh(score[old])  (kept nodes only)
__global__ void pool_gather(const float* __restrict__ h_in, const float* __restrict__ sc,
                            const int* __restrict__ remap, float* __restrict__ h_out,
                            int n_old) {
  long long t = (long long)blockIdx.x * blockDim.x + threadIdx.x;
  if (t >= (long long)n_old * HDIM) return;
  int old = (int)(t >> 7), f = (int)(t & 127);
  int r = remap[old];
  if (r >= 0) h_out[(size_t)r * HDIM + f] = h_in[t] * tanhf(sc[old]);
}

// filter_adj with input emask == 1
__global__ void remap_edges(const int* __restrict__ src, const int* __restrict__ dst,
                            const int* __restrict__ remap, int* __restrict__ src2,
                            int* __restrict__ dst2, float* __restrict__ em2, int ne) {
  int e = blockIdx.x * blockDim.x + threadIdx.x;
  if (e >= ne) return;
  int s = remap[src[e]], d = remap[dst[e]];
  em2[e] = (s >= 0 && d >= 0) ? 1.0f : 0.0f;
  src2[e] = s < 0 ? 0 : s;
  dst2[e] = d < 0 ? 0 : d;
}

__global__ void bn_relu_k(float* __restrict__ h, const float* __restrict__ g,
                          const float* __restrict__ b, const float* __restrict__ m,
                          const float* __restrict__ v, long long n128) {
  long long t = (long long)blockIdx.x * blockDim.x + threadIdx.x;
  if (t >= n128) return;
  int f = (int)(t & 127);
  float y = g[f] * (h[t] - m[f]) * rsqrtf(v[f] + BNEPS) + b[f];
  h[t] = y > 0.0f ? y : 0.0f;
}

// pooled[g][c] = sum_n h[g*n_per + n][c]   (blockDim = 128, grid = BGR)
__global__ __launch_bounds__(128) void graph_sum(const float* __restrict__ h,
                                                 float* __restrict__ pooled, int n_per) {
  int g = blockIdx.x, c = threadIdx.x;
  const float* base = h + (size_t)g * n_per * HDIM + c;
  float s = 0.0f;
  for (int n = 0; n < n_per; ++n) s += base[(size_t)n * HDIM];
  pooled[g * HDIM + c] = s;
}

__global__ void readout(const float* __restrict__ p0, const float* __restrict__ p1,
                        const float* __restrict__ p2, const float* __restrict__ W0,
                        const float* __restrict__ b0, const float* __restrict__ W1,
                        const float* __restrict__ b1, const float* __restrict__ W2,
                        const float* __restrict__ b2, float* __restrict__ out) {
  int t = blockIdx.x * blockDim.x + threadIdx.x;
  if (t >= BGR * DOUT) return;
  int b = t / DOUT, o = t % DOUT;
  float s = b0[o] + b1[o] + b2[o];
  for (int c = 0; c < HDIM; ++c) {
    s += p0[b * HDIM + c] * W0[o * HDIM + c];
    s += p1[b * HDIM + c] * W1[o * HDIM + c];
    s += p2[b * HDIM + c] * W2[o * HDIM + c];
  }
  out[t] = s;
}

// ---------------- launch orchestration ----------------
static inline size_t al256(size_t x) { return (x + 255) & ~(size_t)255; }

extern "C" void kernel_launch(void* const* d_in, const int* in_sizes, int n_in,
                              void* d_out, int out_size, void* d_ws, size_t ws_size,
                              hipStream_t stream) {
  const float* x        = (const float*)d_in[0];
  const int*   eidx     = (const int*)d_in[1];
  const int*   src0     = eidx;
  const int*   dst0     = eidx + ETOT;
  const float* conv1_W  = (const float*)d_in[2];
  const float* conv1_b  = (const float*)d_in[3];
  const float* conv2_W  = (const float*)d_in[4];
  const float* conv2_b  = (const float*)d_in[5];
  const float* score1_W = (const float*)d_in[6];
  const float* score1_b = (const float*)d_in[7];
  const float* score2_W = (const float*)d_in[8];
  const float* score2_b = (const float*)d_in[9];
  const float* bn1_g = (const float*)d_in[10], *bn1_b = (const float*)d_in[11];
  const float* bn1_m = (const float*)d_in[12], *bn1_v = (const float*)d_in[13];
  const float* bn2_g = (const float*)d_in[14], *bn2_b = (const float*)d_in[15];
  const float* bn2_m = (const float*)d_in[16], *bn2_v = (const float*)d_in[17];
  const float* lin0_W = (const float*)d_in[18], *lin0_b = (const float*)d_in[19];
  const float* lin1_W = (const float*)d_in[20], *lin1_b = (const float*)d_in[21];
  const float* lin2_W = (const float*)d_in[22], *lin2_b = (const float*)d_in[23];
  float* out = (float*)d_out;

  // workspace carve-out
  char* p = (char*)d_ws;
  size_t off = 0;
  auto take = [&](size_t bytes) { char* r = p + off; off = al256(off + bytes); return r; };
  float* bufA  = (float*)take((size_t)N0 * HDIM * 4);   // 33.5 MB
  float* bufB  = (float*)take((size_t)N0 * HDIM * 4);   // 33.5 MB
  int*   src2  = (int*)  take((size_t)ETOT * 4);
  int*   dst2  = (int*)  take((size_t)ETOT * 4);
  float* em2   = (float*)take((size_t)ETOT * 4);
  float* deg   = (float*)take((size_t)N0 * 4);
  float* dinv  = (float*)take((size_t)N0 * 4);
  float* shw   = (float*)take((size_t)N0 * 4);
  float* sagg  = (float*)take((size_t)N0 * 4);
  float* sbuf  = (float*)take((size_t)N0 * 4);
  int*   remap = (int*)  take((size_t)N0 * 4);
  float* pool0 = (float*)take((size_t)BGR * HDIM * 4);
  float* pool1 = (float*)take((size_t)BGR * HDIM * 4);
  float* pool2 = (float*)take((size_t)BGR * HDIM * 4);
  (void)ws_size; (void)in_sizes; (void)n_in; (void)out_size;

  const int T = 256;
  auto blk = [&](long long n) { return (unsigned)((n + T - 1) / T); };
  const long long EF = (long long)ETOT * HDIM;

  // ---------- layer 0 (n = N0, 512 nodes/graph) ----------
  gemm_wmma_f32<<<N0 / 128, 256, 0, stream>>>(x, conv1_W, bufA);          // hw0 = x @ W1
  fill_f32<<<blk(N0), T, 0, stream>>>(deg, 1.0f, N0);
  deg_edges<<<blk(ETOT), T, 0, stream>>>(dst0, nullptr, deg, ETOT);
  make_dinv<<<blk(N0), T, 0, stream>>>(deg, dinv, N0);
  fill_f32<<<blk((long long)N0 * HDIM), T, 0, stream>>>(bufB, 0.0f, (long long)N0 * HDIM);
  edge_agg<<<blk(EF), T, 0, stream>>>(src0, dst0, nullptr, dinv, bufA, bufB, ETOT);
  combine<<<blk((long long)N0 * HDIM), T, 0, stream>>>(bufB, bufA, dinv, conv1_b,
                                                       (long long)N0 * HDIM);   // h0 in bufB
  // score 1
  score_gemv<<<blk((long long)N0 * 32), T, 0, stream>>>(bufB, score1_W, shw, N0);
  fill_f32<<<blk(N0), T, 0, stream>>>(sagg, 0.0f, N0);
  score_edge_agg<<<blk(ETOT), T, 0, stream>>>(src0, dst0, nullptr, dinv, shw, sagg, ETOT);
  score_combine<<<blk(N0), T, 0, stream>>>(sbuf, sagg, shw, dinv, score1_b, N0);
  // pool 1: N -> K1
  pool_select<<<BGR, 512, 0, stream>>>(sbuf, NPG, K1, remap);
  pool_gather<<<blk((long long)N0 * HDIM), T, 0, stream>>>(bufB, sbuf, remap, bufA, N0);
  remap_edges<<<blk(ETOT), T, 0, stream>>>(src0, dst0, remap, src2, dst2, em2, ETOT);
  bn_relu_k<<<blk((long long)N1 * HDIM), T, 0, stream>>>(bufA, bn1_g, bn1_b, bn1_m, bn1_v,
                                                         (long long)N1 * HDIM);  // hidden1
  graph_sum<<<BGR, 128, 0, stream>>>(bufA, pool1, K1);

  // ---------- layer 1 (n = N1, 410 nodes/graph) ----------
  gemm_wmma_f32<<<N1 / 128, 256, 0, stream>>>(bufA, conv2_W, bufB);       // hw1 = h1 @ W2
  fill_f32<<<blk(N1), T, 0, stream>>>(deg, 1.0f, N1);
  deg_edges<<<blk(ETOT), T, 0, stream>>>(dst2, em2, deg, ETOT);
  make_dinv<<<blk(N1), T, 0, stream>>>(deg, dinv, N1);
  fill_f32<<<blk((long long)N1 * HDIM), T, 0, stream>>>(bufA, 0.0f, (long long)N1 * HDIM);
  edge_agg<<<blk(EF), T, 0, stream>>>(src2, dst2, em2, dinv, bufB, bufA, ETOT);
  combine<<<blk((long long)N1 * HDIM), T, 0, stream>>>(bufA, bufB, dinv, conv2_b,
                                                       (long long)N1 * HDIM);   // h1 in bufA
  // score 2
  score_gemv<<<blk((long long)N1 * 32), T, 0, stream>>>(bufA, score2_W, shw, N1);
  fill_f32<<<blk(N1), T, 0, stream>>>(sagg, 0.0f, N1);
  score_edge_agg<<<blk(ETOT), T, 0, stream>>>(src2, dst2, em2, dinv, shw, sagg, ETOT);
  score_combine<<<blk(N1), T, 0, stream>>>(sbuf, sagg, shw, dinv, score2_b, N1);
  // pool 2: K1 -> K2 (edges not needed afterwards)
  pool_select<<<BGR, 512, 0, stream>>>(sbuf, K1, K2, remap);
  pool_gather<<<blk((long long)N1 * HDIM), T, 0, stream>>>(bufA, sbuf, remap, bufB, N1);
  bn_relu_k<<<blk((long long)N2 * HDIM), T, 0, stream>>>(bufB, bn2_g, bn2_b, bn2_m, bn2_v,
                                                         (long long)N2 * HDIM);  // hidden2
  graph_sum<<<BGR, 128, 0, stream>>>(bufB, pool2, K2);

  // ---------- readout ----------
  graph_sum<<<BGR, 128, 0, stream>>>(x, pool0, NPG);
  readout<<<blk(BGR * DOUT), T, 0, stream>>>(pool0, pool1, pool2, lin0_W, lin0_b,
                                             lin1_W, lin1_b, lin2_W, lin2_b, out);
}